// ParallelModel_17489106829785
// MI455X (gfx1250) — compile-verified
//
#include <hip/hip_runtime.h>

typedef __attribute__((ext_vector_type(2))) float v2f;
typedef __attribute__((ext_vector_type(8))) float v8f;

#define T_HORIZON 50

__device__ __forceinline__ float fast_tanh(float x) {
#if __has_builtin(__builtin_amdgcn_tanhf)
    return __builtin_amdgcn_tanhf(x);            // native v_tanh_f32 (confirmed)
#elif __has_builtin(__builtin_amdgcn_tanh_f32)
    return __builtin_amdgcn_tanh_f32(x);
#else
    float e = __builtin_amdgcn_exp2f(x * 2.885390081777927f);
    float r = __builtin_amdgcn_rcpf(e + 1.0f);
    return __builtin_fmaf(-2.0f, r, 1.0f);
#endif
}

// half-swap across the 32 lanes: lane l <-> lane l^16
// ds_swizzle group-of-32: offset = xor_mask<<10 | or_mask<<5 | and_mask = 0x401F
__device__ __forceinline__ float xor16(float x) {
    return __int_as_float(__builtin_amdgcn_ds_swizzle(__float_as_int(x), 0x401F));
}

// One wave32 handles 32 batch rows (two 16-row tiles sharing the A weights).
// Hidden-major WMMA: h^T(32x16) = Wh^T(32x8) * x^T(8x16)
//   -> 8x v_wmma_f32_16x16x4_f32 per timestep per wave.
__global__ __launch_bounds__(256) void rollout_wmma(
    const float* __restrict__ u,  const float* __restrict__ y0,
    const float* __restrict__ Wh, const float* __restrict__ bh,
    const float* __restrict__ Wo, const float* __restrict__ bo,
    float* __restrict__ out, int npairs)
{
    const int lane = threadIdx.x & 31;
    const int wv   = threadIdx.x >> 5;
    const int pair = blockIdx.x * 8 + wv;          // 8 waves per 256-thread block
    if (pair >= npairs) return;                    // wave-uniform branch

    const int hi = lane >> 4;
    const int m  = lane & 15;
    const int b0 = pair * 32 + m;                  // tile A batch row
    const int b1 = b0 + 16;                        // tile B batch row

    // ---- A operands: Wh^T.  A[h][c] elem (M=16h+m, Klocal=j+2hi) = Wh[4c+j+2hi][16h+m]
    v2f A00, A01, A10, A11;
    A00.x = Wh[(0 + 2*hi)*32 + 0  + m];  A00.y = Wh[(1 + 2*hi)*32 + 0  + m];
    A01.x = Wh[(4 + 2*hi)*32 + 0  + m];  A01.y = Wh[(5 + 2*hi)*32 + 0  + m];
    A10.x = Wh[(0 + 2*hi)*32 + 16 + m];  A10.y = Wh[(1 + 2*hi)*32 + 16 + m];
    A11.x = Wh[(4 + 2*hi)*32 + 16 + m];  A11.y = Wh[(5 + 2*hi)*32 + 16 + m];

    // ---- bias as C accumulator + Wo in matching D-layout (unit 16h + 8hi + v)
    v8f cb0, cb1, wo0, wo1;
#pragma unroll
    for (int v = 0; v < 8; ++v) {
        cb0[v] = bh[ 0 + 8*hi + v];
        cb1[v] = bh[16 + 8*hi + v];
        wo0[v] = Wo[ 0 + 8*hi + v];
        wo1[v] = Wo[16 + 8*hi + v];
    }
    const float bias_o = bo[0];

    // ---- recurrent state (B-layout): lane<16 -> (y0,y1), lane>=16 -> (y2,y3)
    v2f yi0 = *reinterpret_cast<const v2f*>(y0 + (size_t)b0 * 4 + 2*hi);
    v2f yi1 = *reinterpret_cast<const v2f*>(y0 + (size_t)b1 * 4 + 2*hi);
    float ya0 = yi0.x, yb0 = yi0.y;
    float ya1 = yi1.x, yb1 = yi1.y;

    const float* ub0 = u + (size_t)b0 * (T_HORIZON * 4) + 2*hi;
    const float* ub1 = u + (size_t)b1 * (T_HORIZON * 4) + 2*hi;
    // every lane stores its own batch row: lane l -> batch pair*32 + l
    float* obx = out + (size_t)(pair * 32 + lane) * T_HORIZON;

    v2f bu0 = *reinterpret_cast<const v2f*>(ub0);  // t = 0
    v2f bu1 = *reinterpret_cast<const v2f*>(ub1);

    float pstage = 0.0f;                           // staging for paired b64 store

#pragma unroll 2
    for (int t = 0; t < T_HORIZON; ++t) {
        v2f bu0n = {}, bu1n = {};
        if (t + 1 < T_HORIZON) {
            bu0n = *reinterpret_cast<const v2f*>(ub0 + (t + 1) * 4);
            bu1n = *reinterpret_cast<const v2f*>(ub1 + (t + 1) * 4);
        }
        v2f by0; by0.x = ya0; by0.y = yb0;
        v2f by1; by1.x = ya1; by1.y = yb1;

        // h^T tiles for both batch halves (A, cb shared; 8 WMMAs)
        v8f d0 = __builtin_amdgcn_wmma_f32_16x16x4_f32(false, A00, false, bu0, (short)0, cb0, false, false);
        d0     = __builtin_amdgcn_wmma_f32_16x16x4_f32(false, A01, false, by0, (short)0, d0,  false, false);
        v8f d1 = __builtin_amdgcn_wmma_f32_16x16x4_f32(false, A10, false, bu0, (short)0, cb1, false, false);
        d1     = __builtin_amdgcn_wmma_f32_16x16x4_f32(false, A11, false, by0, (short)0, d1,  false, false);
        v8f d2 = __builtin_amdgcn_wmma_f32_16x16x4_f32(false, A00, false, bu1, (short)0, cb0, false, false);
        d2     = __builtin_amdgcn_wmma_f32_16x16x4_f32(false, A01, false, by1, (short)0, d2,  false, false);
        v8f d3 = __builtin_amdgcn_wmma_f32_16x16x4_f32(false, A10, false, bu1, (short)0, cb1, false, false);
        d3     = __builtin_amdgcn_wmma_f32_16x16x4_f32(false, A11, false, by1, (short)0, d3,  false, false);

        // pred = sum_j tanh(h[b][j]) * Wo[j] ; 4 independent FMA chains, tree combine
        float p0a = 0.0f, p0b = 0.0f, p1a = 0.0f, p1b = 0.0f;
#pragma unroll
        for (int v = 0; v < 8; ++v) {
            p0a = __builtin_fmaf(fast_tanh(d0[v]), wo0[v], p0a);
            p0b = __builtin_fmaf(fast_tanh(d1[v]), wo1[v], p0b);
            p1a = __builtin_fmaf(fast_tanh(d2[v]), wo0[v], p1a);
            p1b = __builtin_fmaf(fast_tanh(d3[v]), wo1[v], p1b);
        }
        float part0 = p0a + p0b;
        float part1 = p1a + p1b;
        float pred0 = part0 + xor16(part0) + bias_o;   // halves hold disjoint hidden units
        float pred1 = part1 + xor16(part1) + bias_o;

        // lane l holds the prediction for batch pair*32 + l after one select:
        float psel = hi ? pred1 : pred0;
        if (t & 1) {
            // full-wave unpredicated b64 store, 8B-aligned ((b*50 + even t)*4)
            v2f s; s.x = pstage; s.y = psel;
            *reinterpret_cast<v2f*>(obx + t - 1) = s;
        } else {
            pstage = psel;
        }

        // state shift y' = [pred, y0, y1, y2]:
        //   yb' = ya ;  ya' = (lane<16) ? pred : old yb of lane-16
        float t0 = xor16(yb0);
        float t1 = xor16(yb1);
        float n0 = hi ? t0 : pred0;
        float n1 = hi ? t1 : pred1;
        yb0 = ya0;  ya0 = n0;
        yb1 = ya1;  ya1 = n1;

        bu0 = bu0n;
        bu1 = bu1n;
    }
}

extern "C" void kernel_launch(void* const* d_in, const int* in_sizes, int n_in,
                              void* d_out, int out_size, void* d_ws, size_t ws_size,
                              hipStream_t stream) {
    const float* u  = (const float*)d_in[0];
    const float* y0 = (const float*)d_in[1];
    const float* Wh = (const float*)d_in[2];
    const float* bh = (const float*)d_in[3];
    const float* Wo = (const float*)d_in[4];
    const float* bo = (const float*)d_in[5];
    float* out = (float*)d_out;

    const int B      = in_sizes[1] / 4;            // y0 is (B, NY=4)
    const int npairs = (B + 31) / 32;              // 32 batch rows per wave
    const int blocks = (npairs + 7) / 8;           // 8 waves per block

    rollout_wmma<<<blocks, 256, 0, stream>>>(u, y0, Wh, bh, Wo, bo, out, npairs);
}